// SalientRegionDetector_86732569576118
// MI455X (gfx1250) — compile-verified
//
#include <hip/hip_runtime.h>

#define BATCH 16
#define HH 384
#define WW 384
#define HW (HH*WW)          // 147456
#define KC 16               // num colors
#define RANK_TGT 117964u    // 0.8*(HW-1), exact in f32
#define PPB 512             // pixels per assign block
#define ABLOCKS (HW/PPB)    // 288

typedef __attribute__((ext_vector_type(2))) float v2f;
typedef __attribute__((ext_vector_type(8))) float v8f;

// ---------- order-preserving float<->uint encode (monotone under unsigned cmp) ----------
__device__ __forceinline__ unsigned encOrd(float f){
  unsigned u = __float_as_uint(f);
  return (u & 0x80000000u) ? ~u : (u | 0x80000000u);
}
__device__ __forceinline__ float decOrd(unsigned u){
  u = (u & 0x80000000u) ? (u & 0x7FFFFFFFu) : ~u;
  return __uint_as_float(u);
}

// XOR-lane min-reduce step via ds_swizzle group-of-32 mode (immediate pattern:
// offset = xor_mask<<10 | and_mask 0x1F). No per-step lane address math.
template<int PAT>
__device__ __forceinline__ unsigned long long swzMin(unsigned long long key){
  int lo = __builtin_amdgcn_ds_swizzle((int)(unsigned)key, PAT);
  int hi = __builtin_amdgcn_ds_swizzle((int)(unsigned)(key >> 32), PAT);
  unsigned long long o = ((unsigned long long)(unsigned)hi << 32) | (unsigned)lo;
  return (o < key) ? o : key;
}

// ---------- grayscale ----------
__global__ void k_gray(const float* __restrict__ x, float* __restrict__ gray){
  int i = blockIdx.x * blockDim.x + threadIdx.x;
  if (i >= BATCH*HW) return;
  int b = i / HW, p = i % HW;
  const float* xb = x + (size_t)b*3*HW;
  gray[i] = 0.299f*xb[p] + 0.587f*xb[HW+p] + 0.114f*xb[2*HW+p];
}

// ---------- sobel magnitude ----------
__global__ void k_sobel(const float* __restrict__ gray, float* __restrict__ edge){
  int p = blockIdx.x * blockDim.x + threadIdx.x;
  int b = blockIdx.y;
  int y = p / WW, xx = p % WW;
  const float* g = gray + (size_t)b*HW;
  float v[3][3];
  #pragma unroll
  for (int dy = 0; dy < 3; dy++){
    #pragma unroll
    for (int dx = 0; dx < 3; dx++){
      int yy = y + dy - 1, xc = xx + dx - 1;
      v[dy][dx] = (yy >= 0 && yy < HH && xc >= 0 && xc < WW) ? g[yy*WW + xc] : 0.f;
    }
  }
  float gx = (v[0][2]-v[0][0]) + 2.f*(v[1][2]-v[1][0]) + (v[2][2]-v[2][0]);
  float gy = (v[2][0]+2.f*v[2][1]+v[2][2]) - (v[0][0]+2.f*v[0][1]+v[0][2]);
  edge[(size_t)b*HW + p] = sqrtf(gx*gx + gy*gy);
}

// ---------- radix select for exact 0.8 quantile (order statistic 117964) ----------
__global__ void k_radix_init(unsigned* __restrict__ hist, unsigned* __restrict__ prefix,
                             unsigned* __restrict__ rnk){
  int i = blockIdx.x * blockDim.x + threadIdx.x;
  if (i < BATCH*256) hist[i] = 0u;
  if (i < BATCH){ prefix[i] = 0u; rnk[i] = RANK_TGT; }
}

__global__ void k_hist(const float* __restrict__ edge, const unsigned* __restrict__ prefix,
                       unsigned* __restrict__ hist, int pass){
  __shared__ unsigned lh[256];
  int t = threadIdx.x;
  lh[t] = 0u;
  __syncthreads();
  int b = blockIdx.y;
  int p = blockIdx.x * 256 + t;
  int shift = 24 - 8*pass;
  unsigned m = pass ? (0xFFFFFFFFu << (shift + 8)) : 0u;
  unsigned u = __float_as_uint(edge[(size_t)b*HW + p]);  // edge >= 0: uint order == float order
  unsigned pf = prefix[b];
  if ((u & m) == (pf & m)) atomicAdd(&lh[(u >> shift) & 0xFFu], 1u);
  __syncthreads();
  if (lh[t]) atomicAdd(&hist[b*256 + t], lh[t]);
}

__global__ void k_select(unsigned* __restrict__ hist, unsigned* __restrict__ prefix,
                         unsigned* __restrict__ rnk, float* __restrict__ high, int pass){
  int b = threadIdx.x;
  if (b >= BATCH) return;
  unsigned r = rnk[b], cum = 0u; int bin = 255;
  for (int j = 0; j < 256; j++){
    unsigned c = hist[b*256 + j];
    if (r < cum + c){ bin = j; break; }
    cum += c;
  }
  rnk[b] = r - cum;
  int shift = 24 - 8*pass;
  prefix[b] |= ((unsigned)bin) << shift;
  if (pass == 3) high[b] = __uint_as_float(prefix[b]);
  for (int j = 0; j < 256; j++) hist[b*256 + j] = 0u;   // clean for next pass
}

// ---------- hysteresis ----------
__global__ void k_hyst_init(const float* __restrict__ edge, const float* __restrict__ high,
                            float* __restrict__ fin){
  int i = blockIdx.x * blockDim.x + threadIdx.x;
  if (i >= BATCH*HW) return;
  int b = i / HW;
  fin[i] = (edge[i] > high[b]) ? 1.f : 0.f;
}

__global__ void k_dilate(const float* __restrict__ fin, const float* __restrict__ edge,
                         const float* __restrict__ high, float* __restrict__ fout){
  int p = blockIdx.x * blockDim.x + threadIdx.x;
  int b = blockIdx.y;
  int y = p / WW, xx = p % WW;
  const float* f = fin + (size_t)b*HW;
  bool neigh = false;
  #pragma unroll
  for (int dy = -1; dy <= 1; dy++){
    #pragma unroll
    for (int dx = -1; dx <= 1; dx++){
      int yy = y + dy, xc = xx + dx;
      if (yy >= 0 && yy < HH && xc >= 0 && xc < WW && f[yy*WW + xc] > 0.f) neigh = true;
    }
  }
  float e  = edge[(size_t)b*HW + p];
  float hi = high[b], lo = 0.5f*hi;
  bool weak = (e <= hi) && (e >= lo);
  float c = f[p];
  fout[(size_t)b*HW + p] = (c > 0.f || (weak && neigh)) ? 1.f : 0.f;
}

// ---------- separable gaussian (radius 5, sigma 1.5) ----------
__device__ __forceinline__ void gaussW(float* w){
  float s = 0.f;
  #pragma unroll
  for (int j = 0; j < 11; j++){ float d = (float)(j-5); w[j] = expf(-(d*d)*(1.f/4.5f)); s += w[j]; }
  float inv = 1.f/s;
  #pragma unroll
  for (int j = 0; j < 11; j++) w[j] *= inv;
}

__global__ void k_gauss_h(const float* __restrict__ in, float* __restrict__ out){
  int p = blockIdx.x * blockDim.x + threadIdx.x;
  int b = blockIdx.y;
  int y = p / WW, xx = p % WW;
  float w[11]; gaussW(w);
  const float* row = in + (size_t)b*HW + (size_t)y*WW;
  float acc = 0.f;
  #pragma unroll
  for (int j = 0; j < 11; j++){
    int xc = xx + j - 5;
    if (xc >= 0 && xc < WW) acc += w[j]*row[xc];
  }
  out[(size_t)b*HW + p] = acc;
}

__global__ void k_gauss_v(const float* __restrict__ in, float* __restrict__ out){
  int p = blockIdx.x * blockDim.x + threadIdx.x;
  int b = blockIdx.y;
  int y = p / WW, xx = p % WW;
  float w[11]; gaussW(w);
  const float* img = in + (size_t)b*HW;
  float acc = 0.f;
  #pragma unroll
  for (int j = 0; j < 11; j++){
    int yy = y + j - 5;
    if (yy >= 0 && yy < HH) acc += w[j]*img[yy*WW + xx];
  }
  out[(size_t)b*HW + p] = acc;
}

// ---------- k-means ----------
__global__ void k_init_centers(const float* __restrict__ x, float* __restrict__ centers){
  int t = threadIdx.x;
  if (t >= BATCH*KC) return;
  int b = t / KC, c = t % KC;
  float step = (float)(HW - 1) / 15.0f;
  int idx = (int)(step * (float)c);
  const float* xb = x + (size_t)b*3*HW;
  centers[t*4+0] = xb[idx];
  centers[t*4+1] = xb[HW + idx];
  centers[t*4+2] = xb[2*HW + idx];
  centers[t*4+3] = 0.f;
}

// WMMA-based assignment + deterministic per-block cluster sums.
// A (16x4 f32): lane<16 holds row M=lane K=0(R),1(G); lane>=16 holds row M=lane-16 K=2(B),3(0).
// B (4x16 f32): lane<16 holds col N=lane K=0,1; lane>=16 holds col N=lane-16 K=2,3.
// D (16x16 f32): vgpr r: lane<16 -> (M=r, N=lane); lane>=16 -> (M=r+8, N=lane-16).
__global__ void k_assign(const float* __restrict__ x, const float* __restrict__ centers,
                         float* __restrict__ part, unsigned char* __restrict__ labels,
                         int writeLabels){
  __shared__ int   slbl[PPB];
  __shared__ float srgb[PPB*3];
  int t = threadIdx.x;
  int wave = t >> 5, lane = t & 31;
  int b = blockIdx.y;
  const float* xr  = x + (size_t)b*3*HW;
  const float* xg  = xr + HW;
  const float* xbl = xr + 2*HW;
  const float* C = centers + b*KC*4;

  int  mn = lane & 15;
  bool lo = lane < 16;
  float c0 = C[mn*4+0], c1 = C[mn*4+1], c2 = C[mn*4+2];
  float cn = c0*c0 + c1*c1 + c2*c2;          // |center_mn|^2 (same for both lane halves)
  v2f Bv = { lo ? c0 : c2, lo ? c1 : 0.f };

  int blockStart = blockIdx.x * PPB;
  #pragma unroll
  for (int k = 0; k < 4; k++){
    int base = blockStart + (wave*4 + k)*16;
    v2f Av = { lo ? xr[base+mn] : xbl[base+mn], lo ? xg[base+mn] : 0.f };
    v8f acc = {};
    acc = __builtin_amdgcn_wmma_f32_16x16x4_f32(false, Av, false, Bv,
                                                (short)0, acc, false, false);
    #pragma unroll
    for (int r = 0; r < 8; r++){
      // distance surrogate: |c|^2 - 2 p.c  (|p|^2 is argmin-invariant)
      float sc = cn - 2.f*acc[r];
      // pack (ordered score, index) into one u64; unsigned min == lexicographic
      // (score, index) min == argmin with first-min tie-break. Fully branchless.
      unsigned long long key = ((unsigned long long)encOrd(sc) << 32) | (unsigned)mn;
      key = swzMin<0x201F>(key);   // xor 8
      key = swzMin<0x101F>(key);   // xor 4
      key = swzMin<0x081F>(key);   // xor 2
      key = swzMin<0x041F>(key);   // xor 1
      int bi = (int)((unsigned)key & 0xFu);
      if (mn == r){                          // one lane per group commits pixel (r / r+8)
        int pl = lo ? r : (r + 8);
        int p  = base + pl;
        int sl = (wave*4 + k)*16 + pl;
        slbl[sl]     = bi;
        srgb[sl*3+0] = xr[p];
        srgb[sl*3+1] = xg[p];
        srgb[sl*3+2] = xbl[p];
        if (writeLabels) labels[(size_t)b*HW + p] = (unsigned char)bi;
      }
    }
  }
  __syncthreads();
  // deterministic fixed-order per-block reduction into partials [cluster][r,g,b,count]
  if (t < KC*4){
    int cl = t >> 2, comp = t & 3;
    float s = 0.f;
    for (int q = 0; q < PPB; q++){
      if (slbl[q] == cl) s += (comp == 3) ? 1.f : srgb[q*3 + comp];
    }
    part[((size_t)b*ABLOCKS + blockIdx.x)*64 + t] = s;
  }
}

__global__ void k_reduce_accum(const float* __restrict__ part, float* __restrict__ accum){
  int t = blockIdx.x * blockDim.x + threadIdx.x;
  if (t >= BATCH*64) return;
  int b = t >> 6, s = t & 63;
  float sum = 0.f;
  for (int j = 0; j < ABLOCKS; j++) sum += part[((size_t)b*ABLOCKS + j)*64 + s];
  accum[t] = sum;
}

__global__ void k_update(float* __restrict__ centers, const float* __restrict__ accum){
  int t = threadIdx.x;
  if (t >= BATCH*KC) return;
  float cnt = accum[t*4+3];
  if (cnt > 0.f){
    float d = fmaxf(cnt, 1.f);
    centers[t*4+0] = accum[t*4+0] / d;
    centers[t*4+1] = accum[t*4+1] / d;
    centers[t*4+2] = accum[t*4+2] / d;
    centers[t*4+3] = 0.f;
  }
}

// ---------- sparsity min/max from cluster counts ----------
__global__ void k_spmm(const float* __restrict__ accum, float* __restrict__ spmm){
  int b = threadIdx.x;
  if (b >= BATCH) return;
  float mn =  3.0e38f, mx = -3.0e38f;
  for (int c = 0; c < KC; c++){
    float cnt = accum[(b*KC + c)*4 + 3];
    if (cnt > 0.f){
      float v = -logf(cnt/(float)HW + 1e-9f);
      mn = fminf(mn, v); mx = fmaxf(mx, v);
    }
  }
  spmm[b*2] = mn; spmm[b*2+1] = mx;
}

// ---------- edge-density min/max ----------
__global__ void k_edmm_init(unsigned* __restrict__ edmm){
  int b = threadIdx.x;
  if (b < BATCH){ edmm[2*b] = 0xFFFFFFFFu; edmm[2*b+1] = 0u; }
}

__global__ void k_edmm(const float* __restrict__ ed, unsigned* __restrict__ edmm){
  __shared__ unsigned smn[256], smx[256];
  int t = threadIdx.x, b = blockIdx.y;
  int p = blockIdx.x * 256 + t;
  unsigned e = encOrd(ed[(size_t)b*HW + p]);
  smn[t] = e; smx[t] = e;
  __syncthreads();
  for (int s = 128; s; s >>= 1){
    if (t < s){ smn[t] = min(smn[t], smn[t+s]); smx[t] = max(smx[t], smx[t+s]); }
    __syncthreads();
  }
  if (t == 0){ atomicMin(&edmm[2*b], smn[0]); atomicMax(&edmm[2*b+1], smx[0]); }
}

// ---------- final combine ----------
__global__ void k_combine(const float* __restrict__ ed, const unsigned char* __restrict__ labels,
                          const float* __restrict__ accum, const float* __restrict__ spmm,
                          const unsigned* __restrict__ edmm,
                          const float* __restrict__ alpha, const float* __restrict__ beta,
                          float* __restrict__ out){
  int i = blockIdx.x * blockDim.x + threadIdx.x;
  if (i >= BATCH*HW) return;
  int b = i / HW;
  float emn = decOrd(edmm[2*b]), emx = decOrd(edmm[2*b+1]);
  float er  = emx - emn; er = (er == 0.f) ? 1.f : er;
  float ne  = (ed[i] - emn) / er;
  int   l   = labels[i];
  float cnt = accum[(b*KC + l)*4 + 3];
  float sp  = -logf(cnt/(float)HW + 1e-9f);
  float smn = spmm[2*b], smx = spmm[2*b+1];
  float sr  = smx - smn; sr = (sr == 0.f) ? 1.f : sr;
  float nc  = (sp - smn) / sr;
  float z   = alpha[0]*ne + beta[0]*nc;
  out[i] = 1.f / (1.f + expf(-z));
}

extern "C" void kernel_launch(void* const* d_in, const int* in_sizes, int n_in,
                              void* d_out, int out_size, void* d_ws, size_t ws_size,
                              hipStream_t stream) {
  (void)in_sizes; (void)n_in; (void)out_size; (void)ws_size;
  const float* x     = (const float*)d_in[0];
  const float* alpha = (const float*)d_in[1];
  const float* beta  = (const float*)d_in[2];
  float* out = (float*)d_out;

  char* ws = (char*)d_ws;
  size_t HWB = (size_t)BATCH*HW*sizeof(float);
  float* gray = (float*)(ws);                 // later reused as edge_density
  float* edge = (float*)(ws + HWB);
  float* fA   = (float*)(ws + 2*HWB);
  float* fB   = (float*)(ws + 3*HWB);         // also gaussian tmp
  char*  sm   = ws + 4*HWB;
  unsigned char* labels = (unsigned char*)sm;           sm += (size_t)BATCH*HW;
  float* part    = (float*)sm;                          sm += (size_t)BATCH*ABLOCKS*64*sizeof(float);
  float* centers = (float*)sm;                          sm += BATCH*KC*4*sizeof(float);
  float* accum   = (float*)sm;                          sm += BATCH*KC*4*sizeof(float);
  unsigned* hist = (unsigned*)sm;                       sm += BATCH*256*sizeof(unsigned);
  unsigned* pfx  = (unsigned*)sm;                       sm += BATCH*sizeof(unsigned);
  unsigned* rnk  = (unsigned*)sm;                       sm += BATCH*sizeof(unsigned);
  float* high    = (float*)sm;                          sm += BATCH*sizeof(float);
  unsigned* edmm = (unsigned*)sm;                       sm += BATCH*2*sizeof(unsigned);
  float* spmm    = (float*)sm;                          sm += BATCH*2*sizeof(float);

  dim3 blk(256);
  dim3 gElem((BATCH*HW + 255)/256);
  dim3 gImg(HW/256, BATCH);      // 576 x 16

  // edge pipeline
  k_gray <<<gElem, blk, 0, stream>>>(x, gray);
  k_sobel<<<gImg,  blk, 0, stream>>>(gray, edge);
  k_radix_init<<<dim3((BATCH*256 + 255)/256), blk, 0, stream>>>(hist, pfx, rnk);
  for (int pass = 0; pass < 4; pass++){
    k_hist  <<<gImg, blk, 0, stream>>>(edge, pfx, hist, pass);
    k_select<<<1, 32, 0, stream>>>(hist, pfx, rnk, high, pass);
  }
  k_hyst_init<<<gElem, blk, 0, stream>>>(edge, high, fA);
  k_dilate<<<gImg, blk, 0, stream>>>(fA, edge, high, fB);
  k_dilate<<<gImg, blk, 0, stream>>>(fB, edge, high, fA);
  k_gauss_h<<<gImg, blk, 0, stream>>>(fA, fB);
  k_gauss_v<<<gImg, blk, 0, stream>>>(fB, gray);     // gray := edge_density

  // k-means (WMMA assignment)
  dim3 gAsg(ABLOCKS, BATCH);     // 288 x 16
  k_init_centers<<<1, 256, 0, stream>>>(x, centers);
  for (int it = 0; it < 10; it++){
    k_assign      <<<gAsg, blk, 0, stream>>>(x, centers, part, labels, 0);
    k_reduce_accum<<<dim3((BATCH*64 + 255)/256), blk, 0, stream>>>(part, accum);
    k_update      <<<1, 256, 0, stream>>>(centers, accum);
  }
  k_assign      <<<gAsg, blk, 0, stream>>>(x, centers, part, labels, 1);
  k_reduce_accum<<<dim3((BATCH*64 + 255)/256), blk, 0, stream>>>(part, accum);

  // normalize + combine
  k_spmm     <<<1, 32, 0, stream>>>(accum, spmm);
  k_edmm_init<<<1, 32, 0, stream>>>(edmm);
  k_edmm     <<<gImg, blk, 0, stream>>>(gray, edmm);
  k_combine  <<<gElem, blk, 0, stream>>>(gray, labels, accum, spmm, edmm, alpha, beta, out);
}